// GNN_68032281968990
// MI455X (gfx1250) — compile-verified
//
#include <hip/hip_runtime.h>

#define NN 50000
#define NE 400000
#define NC 4
#define DD 128
#define NSUB 2048
#define SUBSZ 64

typedef __attribute__((ext_vector_type(2))) float v2f;
typedef __attribute__((ext_vector_type(8))) float v8f;

// Kernel A: y[n,d] = mean_c x[n,c,d] ; also zero the agg accumulator.
__global__ void gnn_mean_zero(const float* __restrict__ x,
                              float* __restrict__ y,
                              float* __restrict__ agg) {
    int i = blockIdx.x * blockDim.x + threadIdx.x;   // over NN * (DD/4)
    if (i >= NN * (DD / 4)) return;
    int n  = i >> 5;            // DD/4 == 32 float4 per row
    int q  = i & 31;            // which float4 within the row
    const float4* xb = (const float4*)(x + (size_t)n * NC * DD) + q;
    float4 a0 = xb[0];
    float4 a1 = xb[DD / 4];
    float4 a2 = xb[2 * (DD / 4)];
    float4 a3 = xb[3 * (DD / 4)];
    float4 r;
    r.x = (a0.x + a1.x + a2.x + a3.x) * 0.25f;
    r.y = (a0.y + a1.y + a2.y + a3.y) * 0.25f;
    r.z = (a0.z + a1.z + a2.z + a3.z) * 0.25f;
    r.w = (a0.w + a1.w + a2.w + a3.w) * 0.25f;
    ((float4*)(y   + (size_t)n * DD))[q] = r;
    ((float4*)(agg + (size_t)n * DD))[q] = make_float4(0.f, 0.f, 0.f, 0.f);
}

// Kernel B: one wave32 per edge; lane l handles 4 consecutive d values.
// agg[dst] += y[src] * w_e  via f32 global atomics (L2-resident target).
__global__ void gnn_scatter(const int* __restrict__ src,
                            const int* __restrict__ dst,
                            const float* __restrict__ ew,
                            const float* __restrict__ y,
                            float* __restrict__ agg) {
    int gid  = blockIdx.x * blockDim.x + threadIdx.x;
    int e    = gid >> 5;
    int lane = threadIdx.x & 31;
    if (e >= NE) return;
    int   s  = src[e];
    int   d  = dst[e];
    float we = ew[e];
    float4 v = ((const float4*)(y + (size_t)s * DD))[lane];
    float* op = agg + (size_t)d * DD + lane * 4;
    atomicAdd(op + 0, v.x * we);
    atomicAdd(op + 1, v.y * we);
    atomicAdd(op + 2, v.z * we);
    atomicAdd(op + 3, v.w * we);
}

// Kernel C: pooled_pre[s,d] = sum_k mask * agg[subG[s,k], d]
__global__ void gnn_pool(const int* __restrict__ subg,
                         const float* __restrict__ agg,
                         float* __restrict__ pooled_pre) {
    int s = blockIdx.x;      // 0..NSUB-1
    int d = threadIdx.x;     // 0..DD-1
    const int* row = subg + (size_t)s * SUBSZ;
    float acc = 0.f;
    #pragma unroll 4
    for (int k = 0; k < SUBSZ; ++k) {
        int idx = row[k];
        if (idx >= 0) acc += agg[(size_t)idx * DD + d];
    }
    pooled_pre[(size_t)s * DD + d] = acc;
}

// Kernel D: pooled[2048,128] = pooled_pre @ W via V_WMMA_F32_16X16X4_F32.
// One wave per 16x16 output tile; K-loop of 32 WMMA steps.
__global__ void gnn_wmma_gemm(const float* __restrict__ A,   // [2048,128]
                              const float* __restrict__ B,   // [128,128] (W)
                              float* __restrict__ C) {       // [2048,128]
    int wid  = (blockIdx.x * blockDim.x + threadIdx.x) >> 5;
    int lane = threadIdx.x & 31;
    int tm = wid >> 3;                 // 0..127  (2048/16 row tiles)
    int tn = wid & 7;                  // 0..7    (128/16 col tiles)

    int m16   = lane & 15;             // row within A tile (ISA A layout)
    int n16   = lane & 15;             // col within B tile (ISA B layout)
    int khalf = (lane >> 4) * 2;       // lanes 0-15 -> K{0,1}; 16-31 -> K{2,3}

    const float* Arow = A + (size_t)(tm * 16 + m16) * DD;
    v8f acc = {};
    #pragma unroll 8
    for (int k = 0; k < DD; k += 4) {
        v2f a, b;
        a.x = Arow[k + khalf + 0];
        a.y = Arow[k + khalf + 1];
        b.x = B[(size_t)(k + khalf + 0) * DD + tn * 16 + n16];
        b.y = B[(size_t)(k + khalf + 1) * DD + tn * 16 + n16];
        acc = __builtin_amdgcn_wmma_f32_16x16x4_f32(
                  /*neg_a=*/false, a, /*neg_b=*/false, b,
                  /*c_mod=*/(short)0, acc, /*reuse_a=*/false, /*reuse_b=*/false);
    }
    // C/D layout: lanes 0-15 -> rows 0..7, lanes 16-31 -> rows 8..15
    int mrow = tm * 16 + (lane >> 4) * 8;
    int ncol = tn * 16 + (lane & 15);
    #pragma unroll
    for (int r = 0; r < 8; ++r)
        C[(size_t)(mrow + r) * DD + ncol] = acc[r];
}

extern "C" void kernel_launch(void* const* d_in, const int* in_sizes, int n_in,
                              void* d_out, int out_size, void* d_ws, size_t ws_size,
                              hipStream_t stream) {
    const float* x    = (const float*)d_in[0];   // [50000,4,128]
    const int*   ei   = (const int*)  d_in[1];   // [2,400000]
    const float* ew   = (const float*)d_in[2];   // [400000]
    const int*   subg = (const int*)  d_in[3];   // [2048,64]
    const float* W    = (const float*)d_in[4];   // [128,128]
    float*       out  = (float*)d_out;           // [2048,128]

    float* y          = (float*)d_ws;                    // 25.6 MB
    float* agg        = y   + (size_t)NN * DD;           // 25.6 MB
    float* pooled_pre = agg + (size_t)NN * DD;           // 1.05 MB

    const int* src = ei;
    const int* dst = ei + NE;

    int nA = NN * (DD / 4);
    gnn_mean_zero<<<(nA + 255) / 256, 256, 0, stream>>>(x, y, agg);
    gnn_scatter<<<(NE * 32 + 255) / 256, 256, 0, stream>>>(src, dst, ew, y, agg);
    gnn_pool<<<NSUB, DD, 0, stream>>>(subg, agg, pooled_pre);
    // 2048/16 * 128/16 = 1024 tiles, 4 waves (128 threads) per block
    gnn_wmma_gemm<<<1024 / 4, 128, 0, stream>>>(pooled_pre, W, out);
}